// GraphAggregation_38319698214926
// MI455X (gfx1250) — compile-verified
//
#include <hip/hip_runtime.h>
#include <stdint.h>

typedef float v2f __attribute__((ext_vector_type(2)));
typedef float v8f __attribute__((ext_vector_type(8)));
typedef unsigned int v4u __attribute__((ext_vector_type(4)));
typedef int v4i __attribute__((ext_vector_type(4)));
typedef int v8i __attribute__((ext_vector_type(8)));

#define BATCH   2
#define N_CTX   512
#define M_TGT   256
#define D_FEAT  32
#define T_LEN   128
#define N_ORDER 2
#define C_OUT   64
#define C_CAT   96

// ---- LDS layout (dword units) --------------------------------------------
// fc' is stored K-pair interleaved: slice element (n, dd) lives at
// (n/2)*64 + dd*2 + (n&1). A B-operand (K,K+1 for one column) is then one
// aligned ds_load_b64, and each half-wave's 16 lanes cover 32 distinct banks
// -> zero conflicts, zero padding.
#define FCP_DW     (N_CTX * D_FEAT)                   // 16384 dwords, contiguous
#define CAT_STRIDE 100                                // 96 data + 4 pad (even, bank-skewed)
#define W_STRIDE   100                                // 96 data + 4 pad (400B: 16B-aligned rows)
#define FCP_OFF    0
#define CAT_OFF    (FCP_OFF + FCP_DW)                 // 16384
#define W_OFF      (CAT_OFF + M_TGT * CAT_STRIDE)     // 41984
#define MV_OFF     (W_OFF + C_OUT * W_STRIDE)         // 48384
#define BIAS_OFF   (MV_OFF + N_CTX)                   // 48896
#define SMEM_DW    (BIAS_OFF + C_OUT)                 // 48960 dwords
#define SMEM_BYTES (SMEM_DW * 4)                      // ~191 KB (< 320 KB WGP LDS)

// ---------------------------------------------------------------------------
// Kernel 1: feat_context [b,n,d,t] -> fc' [b,t, n/2, d, 2] with (1-mask)
// folded in. Coalesced read along t, LDS tile transpose, fully coalesced
// 256B-chunk writes of the pair-interleaved layout.
// ---------------------------------------------------------------------------
__global__ __launch_bounds__(256) void ga_prep_transpose(
    const float* __restrict__ fc,    // [b,n,d,t]
    const float* __restrict__ mask,  // [b,t,n]
    float* __restrict__ fcp)         // [b,t][n/2][d][2]
{
  __shared__ float tile[64 * 131];       // row = dd*2+p, stride 131 (odd skew, conflict-free)
  const int blk = blockIdx.x;            // b * 256 pair-blocks
  const int bi  = blk >> 8;
  const int n2  = blk & 255;
  const int tid = threadIdx.x;

  const float* src = fc + ((size_t)(bi * N_CTX + 2 * n2)) * (D_FEAT * T_LEN);
#pragma unroll
  for (int i = 0; i < 32; ++i) {
    int lin = i * 256 + tid;                         // 8192 = 2*32*128
    int p = lin >> 12, dd = (lin >> 7) & 31, tt = lin & 127;  // tt consecutive: coalesced
    tile[(dd * 2 + p) * 131 + tt] = src[(size_t)(p * D_FEAT + dd) * T_LEN + tt];
  }
  __syncthreads();
#pragma unroll
  for (int i = 0; i < 32; ++i) {
    int lin = i * 256 + tid;
    int tt = lin >> 6, q = lin & 63;                 // q = dd*2+p, consecutive: coalesced
    int p  = q & 1;
    float mk = mask[(size_t)(bi * T_LEN + tt) * N_CTX + 2 * n2 + p];
    fcp[(size_t)(bi * T_LEN + tt) * FCP_DW + n2 * 64 + q] = tile[q * 131 + tt] * (1.0f - mk);
  }
}

// ---------------------------------------------------------------------------
// Kernel 2: one workgroup per (b,t). 16 waves, each owns a 16-row M-tile.
//   Stage A: wave 0 issues ONE tensor_load_to_lds (TDM) for the contiguous
//            64KB fc' slice; other waves concurrently gather feat_target
//            channels and async-copy W (padded rows) via
//            global_load_async_to_lds_b128. TENSORcnt + ASYNCcnt waits, barrier.
//   Stage B: per order: U = adj @ fc' via V_WMMA_F32_16X16X4_F32 (K=512),
//            rowsum folded into K-loop, agg = ft + U*rcp(r+1) -> cat LDS.
//   Stage C: Y = cat[96] @ W^T[96,64] via WMMA (K=96), bias + ReLU, store.
// ---------------------------------------------------------------------------
__global__ __launch_bounds__(512) void ga_main(
    const float* __restrict__ fcp_ws,  // [b,t][n/2][d][2] from prep
    const float* __restrict__ ft_g,    // [b,m,d,t]
    const float* __restrict__ adj,     // [b,2,m,n]
    const float* __restrict__ mask,    // [b,t,n]
    const float* __restrict__ Wm,      // [64,96]
    const float* __restrict__ bias,    // [64]
    float* __restrict__ out)           // [b,m,64,t]
{
  extern __shared__ float smem[];
  float* fcp = smem + FCP_OFF;
  float* cat = smem + CAT_OFF;
  float* Wl  = smem + W_OFF;
  float* mv  = smem + MV_OFF;
  float* bl  = smem + BIAS_OFF;

  const int tid  = threadIdx.x;
  const int bi   = blockIdx.x >> 7;       // / T_LEN
  const int t    = blockIdx.x & (T_LEN - 1);
  const int lane = tid & 31;
  const int wid  = tid >> 5;              // 16 waves
  const int half = lane >> 4;             // K-phase half (fp32 WMMA operand layout)
  const int li   = lane & 15;
  const int m0   = wid * 16;

  // ---- Stage A0: wave 0 fires the Tensor Data Mover for the whole fc' slice.
  if (wid == 0) {
    uint64_t gaddr = (uint64_t)(uintptr_t)fcp_ws
                   + (uint64_t)(bi * T_LEN + t) * (FCP_DW * 4u);
    uint32_t ldsa  = (uint32_t)(uintptr_t)fcp;
    // D# group0: count=1, lds_addr, global_addr[56:0], type=2 ("image")
    v4u g0 = { 1u, ldsa, (uint32_t)gaddr,
               (uint32_t)((gaddr >> 32) & 0x01FFFFFFu) | 0x80000000u };
    // D# group1: data_size=4B(code2); 1 line of 16384 dwords (64KB burst);
    //            tensor_dim0=16384, tensor_dim1=1, tile_dim0=16384, tile_dim1=1
    v8i g1 = { 0x00020000, 0x40000000, 0x00010000, 0x40000000,
               1, 16384, 0, 0 };
    // D# groups 2/3: higher dims = 1 (single iteration), strides 0
    v4i g2 = { 1, 1, 0, 0x00010000 };
    v4i g3 = { 0, 0x00010000, 0x00010000, 0 };
    v8i g4 = { 0, 0, 0, 0, 0, 0, 0, 0 };   // 6-arg toolchain: zero-filled extra group
    __builtin_amdgcn_tensor_load_to_lds(g0, g1, g2, g3, g4, 0);
  }

  // ---- Stage A1: async-copy W into padded LDS rows (ASYNCcnt path).
  {
    const uint64_t wbase = (uint64_t)(uintptr_t)Wm;
    const uint32_t lds_w = (uint32_t)(uintptr_t)Wl;
#pragma unroll
    for (int j = 0; j < 3; ++j) {
      int c  = tid + 512 * j;                 // 1536 x 16B chunks (64 rows x 24)
      int o  = c / 24, r4 = c - o * 24;
      uint32_t ldst = lds_w + (uint32_t)(o * (W_STRIDE * 4) + r4 * 16);
      uint32_t voff = (uint32_t)(o * (C_CAT * 4) + r4 * 16);
      asm volatile("global_load_async_to_lds_b128 %0, %1, %2"
                   :: "v"(ldst), "v"(voff), "s"(wbase) : "memory");
    }
  }

  // ---- Stage A2: cooperative staging of the rest (overlaps TDM + async).
  {
    // feat_target t-slice -> cat channels [0,32)
    const float* ftb = ft_g + (size_t)bi * M_TGT * D_FEAT * T_LEN + t;
#pragma unroll
    for (int j = 0; j < 16; ++j) {
      int lin = j * 512 + tid;
      int row = lin >> 5, dd = lin & 31;
      cat[row * CAT_STRIDE + dd] = ftb[(size_t)(row * D_FEAT + dd) * T_LEN];
    }
    if (tid < N_CTX) mv[tid] = 1.0f - mask[(size_t)(bi * T_LEN + t) * N_CTX + tid];
    if (tid < C_OUT) bl[tid] = bias[tid];
  }

  asm volatile("s_wait_asynccnt 0" ::: "memory");
  if (wid == 0) __builtin_amdgcn_s_wait_tensorcnt(0);
  __syncthreads();

  // ---- Stage B: adjacency GEMMs, order 0/1.
  for (int ord = 0; ord < N_ORDER; ++ord) {
    const float* arow = adj + ((size_t)(bi * N_ORDER + ord) * M_TGT + (m0 + li)) * N_CTX
                            + 2 * half;                  // per-lane A row, K-phase shifted
    __builtin_prefetch(arow, 0, 0);                       // global_prefetch_b8
    v8f acc0 = {};  // D columns 0..15
    v8f acc1 = {};  // D columns 16..31
    float rsum = 0.0f;
#pragma unroll 4
    for (int k = 0; k < N_CTX; k += 4) {
      // A 16x4 fp32 tile: lane<16 holds K={k,k+1}, lane>=16 holds K={k+2,k+3}
      v2f a2 = *(const v2f*)(arow + k);
      // fold rowsum r = adj @ (1-mask) into the same K loop
      v2f m2 = *(const v2f*)(mv + k + 2 * half);
      rsum += a2.x * m2.x + a2.y * m2.y;
      // B tiles: pair-interleaved fc' -> single ds_load_b64, all 64 banks hit once
      const float* frow = fcp + ((k >> 1) + half) * 64;
      v2f b0 = *(const v2f*)(frow + 2 * li);
      acc0 = __builtin_amdgcn_wmma_f32_16x16x4_f32(false, a2, false, b0, (short)0, acc0, false, false);
      v2f b1 = *(const v2f*)(frow + 32 + 2 * li);
      acc1 = __builtin_amdgcn_wmma_f32_16x16x4_f32(false, a2, false, b1, (short)0, acc1, false, false);
    }
    // combine K-phase halves: lane j / j+16 hold row m0+(j&15) partials
    float rtot = rsum + __shfl_xor(rsum, 16, 32);
    // scale + residual into cat channels [32+32*ord, 64+32*ord)
#pragma unroll
    for (int r = 0; r < 8; ++r) {
      int rowrel = r + 8 * half;                       // D layout: M = r + 8*(lane>=16)
      float rs   = __shfl(rtot, rowrel, 32);           // ds_bpermute gather of row sum
      float inv  = __builtin_amdgcn_rcpf(rs + 1.0f);   // v_rcp_f32 (normalization weight)
      int row    = m0 + rowrel;
      float ft0  = cat[row * CAT_STRIDE + li];
      float ft1  = cat[row * CAT_STRIDE + 16 + li];
      cat[row * CAT_STRIDE + 32 + 32 * ord + li]      = ft0 + acc0[r] * inv;
      cat[row * CAT_STRIDE + 32 + 32 * ord + 16 + li] = ft1 + acc1[r] * inv;
    }
  }
  // No barrier: each wave's stage-C A-tile reads only its own cat rows (LDS in-order per wave).

  // ---- Stage C: Y[16,64] = cat[16,96] @ W^T[96,64]
  v8f ys[4] = {v8f{}, v8f{}, v8f{}, v8f{}};
#pragma unroll 4
  for (int k = 0; k < C_CAT; k += 4) {
    v2f a2 = *(const v2f*)(cat + (m0 + li) * CAT_STRIDE + k + 2 * half);
#pragma unroll
    for (int nt = 0; nt < 4; ++nt) {
      v2f b2 = *(const v2f*)(Wl + (nt * 16 + li) * W_STRIDE + k + 2 * half);
      ys[nt] = __builtin_amdgcn_wmma_f32_16x16x4_f32(false, a2, false, b2, (short)0, ys[nt], false, false);
    }
  }

  // ---- Epilogue: bias + ReLU, scatter to out[b,m,o,t]
  float* outb = out + (size_t)bi * M_TGT * C_OUT * T_LEN + t;
#pragma unroll
  for (int nt = 0; nt < 4; ++nt) {
    int o = nt * 16 + li;
    float bv = bl[o];
#pragma unroll
    for (int r = 0; r < 8; ++r) {
      int row = m0 + r + 8 * half;
      float v = ys[nt][r] + bv;
      v = v > 0.0f ? v : 0.0f;
      outb[(size_t)(row * C_OUT + o) * T_LEN] = v;
    }
  }
}

// ---------------------------------------------------------------------------
extern "C" void kernel_launch(void* const* d_in, const int* in_sizes, int n_in,
                              void* d_out, int out_size, void* d_ws, size_t ws_size,
                              hipStream_t stream) {
  const float* feat_context = (const float*)d_in[0];  // [2,512,32,128]
  const float* feat_target  = (const float*)d_in[1];  // [2,256,32,128]
  const float* adj          = (const float*)d_in[2];  // [2,2,256,512]
  const float* mask         = (const float*)d_in[3];  // [2,128,512]
  const float* W_mlp        = (const float*)d_in[4];  // [64,96]
  const float* b_mlp        = (const float*)d_in[5];  // [64]
  float* out    = (float*)d_out;                      // [2,256,64,128]
  float* fcp_ws = (float*)d_ws;                       // [2,128][256][32][2] fp32 (~16.8 MB)

  ga_prep_transpose<<<dim3(BATCH * (N_CTX / 2)), dim3(256), 0, stream>>>(
      feat_context, mask, fcp_ws);
  ga_main<<<dim3(BATCH * T_LEN), dim3(512), SMEM_BYTES, stream>>>(
      fcp_ws, feat_target, adj, mask, W_mlp, b_mlp, out);
}